// L1TV2DLayer_23536420782356
// MI455X (gfx1250) — compile-verified
//
#include <hip/hip_runtime.h>
#include <cstdint>
#include <cstddef>

// Problem geometry (fixed by the reference).
#define NB 8
#define CC 3
#define HH 256
#define WW 256
#define HW (HH * WW)
#define S_ELEMS (NB * CC * HH * WW)   // 1,572,864 floats
#define NUM_ITER 4

// Tiling: one wave32 block processes 32 independent lines (rows or cols).
#define LINES 32
#define NLINE_BLOCKS ((NB * CC * HH) / LINES)  // 192 (same count for cols: 24 imgs * 8 strips)
#define ROW_STRIDE 257   // LDS floats per row:  bank (r + i) % 64  -> conflict free
#define COL_STRIDE 33    // LDS floats per k-row (transposed col tile): bank (33k + j) % 64

typedef unsigned v4u __attribute__((ext_vector_type(4)));
typedef unsigned v8u __attribute__((ext_vector_type(8)));

// ---------------------------------------------------------------------------
// Tensor Data Mover: one-wave 2D tile DMA global->LDS with hardware padding.
// D# encoding per CDNA5 ISA:
//   group0: [1:0]=count=1, [63:32]=lds_addr(bytes), [120:64]=global_addr,
//           [127:126]=type=2
//   group1: [17:16]=data_size(2->4B), [20]=pad_enable, [24:22]=pad_interval
//           (4->32dw, 7->256dw), [31:25]=pad_amount(0->1dw),
//           [79:48]=tensor_dim0, [111:80]=tensor_dim1, [127:112]=tile_dim0,
//           [143:128]=tile_dim1, [207:160]=tensor_dim0_stride
// groups 2/3 zero (2D tile, tile_dim2=0 => unused).
// ---------------------------------------------------------------------------
__device__ __forceinline__ void tdm_load_2d(unsigned lds_byte_off, const void* gptr,
                                            unsigned tile_d0_dw, unsigned tile_d1,
                                            unsigned row_stride_dw,
                                            unsigned pad_interval_code) {
    const unsigned long long ga = (unsigned long long)(uintptr_t)gptr;
    v4u g0;
    g0[0] = 1u;                                             // count=1 (user descriptor)
    g0[1] = lds_byte_off;                                   // LDS dest (bytes)
    g0[2] = (unsigned)(ga & 0xFFFFFFFFull);                 // global_addr[31:0]
    g0[3] = (unsigned)((ga >> 32) & 0x01FFFFFFull)          // global_addr[56:32]
            | 0x80000000u;                                  // type=2 ("image")
    v8u g1;
    g1[0] = (2u << 16) | (1u << 20) | (pad_interval_code << 22); // 4B, pad_en, interval, amt=1dw
    g1[1] = (tile_d0_dw & 0xFFFFu) << 16;                   // tensor_dim0 lo16 (== tile_d0)
    g1[2] = ((tile_d0_dw >> 16) & 0xFFFFu) | ((tile_d1 & 0xFFFFu) << 16); // td0 hi | td1 lo
    g1[3] = ((tile_d1 >> 16) & 0xFFFFu) | (tile_d0_dw << 16);             // td1 hi | tile_dim0
    g1[4] = tile_d1;                                        // tile_dim1 (tile_dim2=0)
    g1[5] = row_stride_dw;                                  // tensor_dim0_stride lo32
    g1[6] = 0u;                                             // stride hi16, dim1_stride lo16
    g1[7] = 0u;                                             // dim1_stride hi32
    const v4u gz = {0u, 0u, 0u, 0u};
    asm volatile("tensor_load_to_lds %0, %1, %2, %3"
                 :: "s"(g0), "s"(g1), "s"(gz), "s"(gz) : "memory");
}

// ---------------------------------------------------------------------------
// Exact 1D TV prox (Condat 2013), faithful to the reference's branch structure,
// run in-place on an LDS-resident line with compile-time element stride.
// In-place is safe: every y-read after a segment finalizes is at an index
// strictly greater than that segment's last written position.
// ---------------------------------------------------------------------------
template <int STR>
__device__ __forceinline__ void tv1d_condat(float* x, float lam) {
    const int N = 256;
    int k = 0, k0 = 0, km = 0, kp = 0;
    float y0 = x[0];
    float vmin = y0 - lam, vmax = y0 + lam;
    float umin = lam, umax = -lam;
    bool done = false;
    int it = 0;
    const int max_iter = 8 * N;   // same safety cap as the reference
    while (!done && it < max_iter) {
        ++it;
        const bool in_main = k < N - 1;
        const int knext = in_main ? (k + 1) : (N - 1);
        const float ynext = x[knext * STR];
        const bool neg = (ynext + umin) < (vmin - lam);
        const bool pos = (ynext + umax) > (vmax + lam);
        const bool c3 = in_main && neg;
        const bool c4 = in_main && !neg && pos;
        const bool c5 = in_main && !neg && !pos;
        const bool c7 = !in_main && (umin < 0.0f);
        const bool c8 = !in_main && (umin >= 0.0f) && (umax > 0.0f);
        const bool c9 = !in_main && (umin >= 0.0f) && (umax <= 0.0f);

        if (c5) {
            const int k5 = k + 1;
            float umin5 = umin + ynext - vmin;
            float umax5 = umax + ynext - vmax;
            const float len5 = (float)(k5 - k0 + 1);
            if (umin5 >= lam)  { vmin = vmin + (umin5 - lam) / len5; umin5 = lam;  km = k5; }
            if (umax5 <= -lam) { vmax = vmax + (umax5 + lam) / len5; umax5 = -lam; kp = k5; }
            umin = umin5; umax = umax5; k = k5;
        } else {
            const bool lo = c3 || c7;
            const bool hi = c4 || c8;
            const int seg_hi = lo ? km : (hi ? kp : (N - 1));
            float seg_val;
            if (lo)      seg_val = vmin;
            else if (hi) seg_val = vmax;
            else { const float seg_len = (float)(k - k0 + 1); seg_val = vmin + umin / seg_len; }
            int kr = (lo ? km : kp) + 1;
            if (kr > N - 1) kr = N - 1;
            const float ykr = x[kr * STR];     // read BEFORE segment write (in-place safety)
            for (int i = k0; i <= seg_hi; ++i) x[i * STR] = seg_val;
            if (c9) {
                done = true;
            } else {
                const float vmin_n = (c3 || c7) ? ykr : (c4 ? (ykr - 2.0f * lam) : vmin);
                const float vmax_n = c3 ? (ykr + 2.0f * lam) : ((c4 || c8) ? ykr : vmax);
                const float umin_n = (c3 || c4 || c7) ? lam  : (c8 ? (ykr - lam - vmin) : umin);
                const float umax_n = (c3 || c4 || c8) ? -lam : (c7 ? (ykr + lam - vmax) : umax);
                const int km_n = (c3 || c4 || c7) ? kr : km;
                const int kp_n = (c3 || c4 || c8) ? kr : kp;
                k = kr; k0 = kr; km = km_n; kp = kp_n;
                vmin = vmin_n; vmax = vmax_n; umin = umin_n; umax = umax_n;
                if ((c7 || c8) && (kr == N - 1)) {     // 'fin'
                    x[(N - 1) * STR] = vmin + umin;
                    done = true;
                }
            }
        }
    }
}

// ---------------------------------------------------------------------------
// Elementwise helpers (float4, L2-resident traffic).
// ---------------------------------------------------------------------------
__global__ void __launch_bounds__(256) zero_kernel(float4* p, int n4) {
    int i = blockIdx.x * 256 + threadIdx.x;
    if (i < n4) { float4 z; z.x = 0.f; z.y = 0.f; z.z = 0.f; z.w = 0.f; p[i] = z; }
}

__global__ void __launch_bounds__(256) add_kernel(const float4* a, const float4* b,
                                                  float4* o, int n4) {
    int i = blockIdx.x * 256 + threadIdx.x;
    if (i < n4) {
        float4 av = a[i], bv = b[i], ov;
        ov.x = av.x + bv.x; ov.y = av.y + bv.y; ov.z = av.z + bv.z; ov.w = av.w + bv.w;
        o[i] = ov;
    }
}

// ---------------------------------------------------------------------------
// Row prox, TDM staging: TZ holds T=X+P on entry; on exit TZ holds Z and
// P holds T - Z.  One tensor_load_to_lds per wave; hardware pad -> stride 257.
// ---------------------------------------------------------------------------
__global__ void __launch_bounds__(32) tv_row_tdm(float* TZ, float* P, const float* lmbd) {
    __shared__ float tile[LINES * ROW_STRIDE];      // 32,896 B
    const int tid = threadIdx.x;
    const int row0 = blockIdx.x * LINES;
    const int img = row0 / HH;
    const float lam = expf(lmbd[img % CC]);
    float* gbase = TZ + (size_t)row0 * WW;

    // 32 rows x 256 dw, row stride 256 dw, +1 dw pad every 256 dw (code 7).
    tdm_load_2d(0u, gbase, 256u, (unsigned)LINES, 256u, 7u);
    __builtin_amdgcn_s_wait_tensorcnt(0);
    __syncthreads();

    tv1d_condat<1>(&tile[tid * ROW_STRIDE], lam);
    __syncthreads();

    float* pbase = P + (size_t)row0 * WW;
    for (int c = tid; c < 2048; c += 32) {
        const int f = c * 4;
        const int r = f >> 8;
        const float* ls = &tile[f + r];
        float4 t4 = *(const float4*)(gbase + f);   // re-read T (L2 resident)
        float4 z4, p4;
        z4.x = ls[0]; z4.y = ls[1]; z4.z = ls[2]; z4.w = ls[3];
        p4.x = t4.x - z4.x; p4.y = t4.y - z4.y; p4.z = t4.z - z4.z; p4.w = t4.w - z4.w;
        *(float4*)(gbase + f) = z4;                // Z overwrites T
        *(float4*)(pbase + f) = p4;
    }
}

// ---------------------------------------------------------------------------
// Column prox, TDM staging: U holds Z+Q; writes X (d_out) and Q = U - X.
// Tile = 32-col strip, 256 k-rows; hardware pad -> transposed stride-33 layout.
// ---------------------------------------------------------------------------
__global__ void __launch_bounds__(32) tv_col_tdm(const float* U, float* X, float* Q,
                                                 const float* lmbd) {
    __shared__ float tile[HH * COL_STRIDE];         // 33,792 B
    const int tid = threadIdx.x;
    const int img = blockIdx.x >> 3;
    const int c0 = (blockIdx.x & 7) * LINES;
    const float lam = expf(lmbd[img % CC]);
    const float* gbase = U + (size_t)img * HW + c0;

    // 256 rows x 32 dw, row stride 256 dw, +1 dw pad every 32 dw (code 4).
    tdm_load_2d(0u, gbase, 32u, (unsigned)HH, 256u, 4u);
    __builtin_amdgcn_s_wait_tensorcnt(0);
    __syncthreads();

    tv1d_condat<COL_STRIDE>(&tile[tid], lam);       // thread tid walks column c0+tid
    __syncthreads();

    float* xbase = X + (size_t)img * HW + c0;
    float* qbase = Q + (size_t)img * HW + c0;
    for (int k = 0; k < HH; ++k) {                  // coalesced across the wave
        const float xv = tile[k * COL_STRIDE + tid];
        const float uv = gbase[(size_t)k * WW + tid];
        xbase[(size_t)k * WW + tid] = xv;
        qbase[(size_t)k * WW + tid] = uv - xv;
    }
}

// ---------------------------------------------------------------------------
// Per-lane async-to-LDS variants (kept compiled; ASYNCcnt path).
// ---------------------------------------------------------------------------
__global__ void __launch_bounds__(32) tv_row_async(float* TZ, float* P, const float* lmbd) {
    __shared__ float tile[LINES * ROW_STRIDE];
    const int tid = threadIdx.x;
    const int row0 = blockIdx.x * LINES;
    const int img = row0 / HH;
    const float lam = expf(lmbd[img % CC]);
    float* gbase = TZ + (size_t)row0 * WW;

    for (int c = tid; c < 2048; c += 32) {
        const int f = c * 4;
        const int r = f >> 8;
        const unsigned lds_off = (unsigned)(f + r) * 4u;
        const unsigned long long ga = (unsigned long long)(uintptr_t)(gbase + f);
        asm volatile("global_load_async_to_lds_b128 %0, %1, off"
                     :: "v"(lds_off), "v"(ga) : "memory");
    }
    asm volatile("s_wait_asynccnt 0" ::: "memory");
    __syncthreads();
    tv1d_condat<1>(&tile[tid * ROW_STRIDE], lam);
    __syncthreads();

    float* pbase = P + (size_t)row0 * WW;
    for (int c = tid; c < 2048; c += 32) {
        const int f = c * 4;
        const int r = f >> 8;
        const float* ls = &tile[f + r];
        float4 t4 = *(const float4*)(gbase + f);
        float4 z4, p4;
        z4.x = ls[0]; z4.y = ls[1]; z4.z = ls[2]; z4.w = ls[3];
        p4.x = t4.x - z4.x; p4.y = t4.y - z4.y; p4.z = t4.z - z4.z; p4.w = t4.w - z4.w;
        *(float4*)(gbase + f) = z4;
        *(float4*)(pbase + f) = p4;
    }
}

__global__ void __launch_bounds__(32) tv_col_async(const float* U, float* X, float* Q,
                                                   const float* lmbd) {
    __shared__ float tile[HH * COL_STRIDE];
    const int tid = threadIdx.x;
    const int img = blockIdx.x >> 3;
    const int c0 = (blockIdx.x & 7) * LINES;
    const float lam = expf(lmbd[img % CC]);
    const float* gbase = U + (size_t)img * HW + c0;

    for (int c = tid; c < 2048; c += 32) {
        const int k = c >> 3;
        const int p = c & 7;
        const unsigned lds_off = (unsigned)(k * COL_STRIDE + p * 4) * 4u;
        const unsigned long long ga =
            (unsigned long long)(uintptr_t)(gbase + (size_t)k * WW + p * 4);
        asm volatile("global_load_async_to_lds_b128 %0, %1, off"
                     :: "v"(lds_off), "v"(ga) : "memory");
    }
    asm volatile("s_wait_asynccnt 0" ::: "memory");
    __syncthreads();
    tv1d_condat<COL_STRIDE>(&tile[tid], lam);
    __syncthreads();

    float* xbase = X + (size_t)img * HW + c0;
    float* qbase = Q + (size_t)img * HW + c0;
    for (int k = 0; k < HH; ++k) {
        const float xv = tile[k * COL_STRIDE + tid];
        const float uv = gbase[(size_t)k * WW + tid];
        xbase[(size_t)k * WW + tid] = xv;
        qbase[(size_t)k * WW + tid] = uv - xv;
    }
}

// ---------------------------------------------------------------------------
// Fused fallbacks (used only if the workspace cannot hold the T buffer).
// ---------------------------------------------------------------------------
__global__ void __launch_bounds__(32) tv_row_fused(const float* A, float* P, float* Zout,
                                                   const float* lmbd) {
    __shared__ float tile[LINES * ROW_STRIDE];
    const int tid = threadIdx.x;
    const int row0 = blockIdx.x * LINES;
    const int img = row0 / HH;
    const float lam = expf(lmbd[img % CC]);
    const float* abase = A + (size_t)row0 * WW;
    float* pbase = P + (size_t)row0 * WW;
    float* zbase = Zout + (size_t)row0 * WW;

    for (int c = tid; c < 2048; c += 32) {
        const int f = c * 4, r = f >> 8;
        float4 a4 = *(const float4*)(abase + f);
        float4 p4 = *(const float4*)(pbase + f);
        float* ls = &tile[f + r];
        ls[0] = a4.x + p4.x; ls[1] = a4.y + p4.y; ls[2] = a4.z + p4.z; ls[3] = a4.w + p4.w;
    }
    __syncthreads();
    tv1d_condat<1>(&tile[tid * ROW_STRIDE], lam);
    __syncthreads();
    for (int c = tid; c < 2048; c += 32) {
        const int f = c * 4, r = f >> 8;
        float4 a4 = *(const float4*)(abase + f);
        float4 p4 = *(const float4*)(pbase + f);
        const float* ls = &tile[f + r];
        float4 z4, np4;
        z4.x = ls[0]; z4.y = ls[1]; z4.z = ls[2]; z4.w = ls[3];
        np4.x = (a4.x + p4.x) - z4.x; np4.y = (a4.y + p4.y) - z4.y;
        np4.z = (a4.z + p4.z) - z4.z; np4.w = (a4.w + p4.w) - z4.w;
        *(float4*)(zbase + f) = z4;
        *(float4*)(pbase + f) = np4;
    }
}

__global__ void __launch_bounds__(32) tv_col_fused(float* XZ, float* Q, const float* lmbd) {
    __shared__ float tile[HH * COL_STRIDE];
    const int tid = threadIdx.x;
    const int img = blockIdx.x >> 3;
    const int c0 = (blockIdx.x & 7) * LINES;
    const float lam = expf(lmbd[img % CC]);
    float* zb = XZ + (size_t)img * HW + c0;
    float* qb = Q + (size_t)img * HW + c0;

    for (int c = tid; c < 2048; c += 32) {
        const int k = c >> 3, p = c & 7;
        float4 z4 = *(const float4*)(zb + (size_t)k * WW + p * 4);
        float4 q4 = *(const float4*)(qb + (size_t)k * WW + p * 4);
        float* ls = &tile[k * COL_STRIDE + p * 4];
        ls[0] = z4.x + q4.x; ls[1] = z4.y + q4.y; ls[2] = z4.z + q4.z; ls[3] = z4.w + q4.w;
    }
    __syncthreads();
    tv1d_condat<COL_STRIDE>(&tile[tid], lam);
    __syncthreads();
    for (int k = 0; k < HH; ++k) {
        const float u = zb[(size_t)k * WW + tid] + qb[(size_t)k * WW + tid];
        const float xv = tile[k * COL_STRIDE + tid];
        zb[(size_t)k * WW + tid] = xv;          // X overwrites Z
        qb[(size_t)k * WW + tid] = u - xv;
    }
}

// ---------------------------------------------------------------------------
// Host orchestration: init P=Q=0, then 4 Dykstra iterations.
// ---------------------------------------------------------------------------
extern "C" void kernel_launch(void* const* d_in, const int* in_sizes, int n_in,
                              void* d_out, int out_size, void* d_ws, size_t ws_size,
                              hipStream_t stream) {
    (void)in_sizes; (void)n_in; (void)out_size;
    const float* y    = (const float*)d_in[0];
    const float* lmbd = (const float*)d_in[1];
    float* X  = (float*)d_out;
    float* ws = (float*)d_ws;
    const size_t S = (size_t)S_ELEMS;
    float* P = ws;
    float* Q = ws + S;
    float* T = ws + 2 * S;
    const bool tdmPath = ws_size >= 3 * S * sizeof(float);

    {   // P = Q = 0 (contiguous 2*S floats at ws start)
        const int n4 = (int)((2 * S) / 4);
        zero_kernel<<<(n4 + 255) / 256, 256, 0, stream>>>((float4*)ws, n4);
    }
    const int n4 = (int)(S / 4);

    for (int it = 0; it < NUM_ITER; ++it) {
        const float* Xin = (it == 0) ? y : (const float*)X;
        if (tdmPath) {
            add_kernel<<<(n4 + 255) / 256, 256, 0, stream>>>(
                (const float4*)Xin, (const float4*)P, (float4*)T, n4);      // T = X + P
            tv_row_tdm<<<NLINE_BLOCKS, 32, 0, stream>>>(T, P, lmbd);        // T->Z, P=T-Z
            add_kernel<<<(n4 + 255) / 256, 256, 0, stream>>>(
                (const float4*)T, (const float4*)Q, (float4*)T, n4);        // T = Z + Q
            tv_col_tdm<<<NLINE_BLOCKS, 32, 0, stream>>>(T, X, Q, lmbd);     // X, Q=U-X
        } else {
            tv_row_fused<<<NLINE_BLOCKS, 32, 0, stream>>>(Xin, P, X, lmbd); // X holds Z
            tv_col_fused<<<NLINE_BLOCKS, 32, 0, stream>>>(X, Q, lmbd);      // X holds X
        }
    }
}